// LinOSSLayer_69698729279782
// MI455X (gfx1250) — compile-verified
//
#include <hip/hip_runtime.h>
#include <math.h>

typedef float v2f __attribute__((ext_vector_type(2)));
typedef float v8f __attribute__((ext_vector_type(8)));

#define Q       256
#define NSEQ    4096
#define BATCH   16
#define MTOT    (BATCH * NSEQ)   // 65536 rows
#define PAIRSTR 160              // dwords per k-pair row: >=128, ==32 mod 64 (bank split)

// LDS layout: sW[(k>>1)*PAIRSTR + 2*p_local + (k&1)] = W[p0+p_local][k]
// -> B fragment (rows k,k+1 @ col p) is one aligned ds_load_b64.
__device__ __forceinline__ int wofs(int k, int p) {
    return (k >> 1) * PAIRSTR + 2 * p + (k & 1);
}

// ---------------------------------------------------------------------------
// GEMM 1:  Bu[m][p] = sum_k U[m][k] * W_B[p][k] + b_B[p]
// V_WMMA_F32_16X16X4_F32.  Wave = 32 rows x 64 cols (2x4 tiles).
// Block = 8 waves = 256 rows x 64 cols.  grid = (MTOT/256, Q/64).
// ---------------------------------------------------------------------------
__global__ __launch_bounds__(256) void gemm_bu_kernel(
    const float* __restrict__ U,    // (MTOT, Q)
    const float* __restrict__ W,    // (Q, Q) p-major (original layout)
    const float* __restrict__ bB,   // (Q)
    float* __restrict__ Bu)         // (MTOT, Q)
{
    __shared__ float sW[(Q / 2) * PAIRSTR];   // 80 KB

    const int tid  = threadIdx.x;
    const int lane = tid & 31;
    const int wave = tid >> 5;
    const int m0   = (blockIdx.x * 8 + wave) * 32;
    const int p0   = blockIdx.y * 64;

    // cooperative transposed stage: thread t covers k=t, iterate p_local
    for (int i = 0; i < 64; ++i)
        sW[wofs(tid, i)] = W[(size_t)(p0 + i) * Q + tid];
    __syncthreads();

    const int l15   = lane & 15;
    const int khalf = (lane >> 4) * 2;   // 0 | 2

    v8f acc[2][4];
#pragma unroll
    for (int mi = 0; mi < 2; ++mi)
#pragma unroll
        for (int pi = 0; pi < 4; ++pi) acc[mi][pi] = (v8f){};

    const float* arow0 = U + (size_t)(m0 + l15) * Q + khalf;
    const float* arow1 = arow0 + (size_t)16 * Q;

    for (int k0 = 0; k0 < Q; k0 += 4) {
        v2f a0 = *(const v2f*)(arow0 + k0);
        v2f a1 = *(const v2f*)(arow1 + k0);
        const int base = wofs(k0 + khalf, l15);   // even -> 8B aligned
#pragma unroll
        for (int pi = 0; pi < 4; ++pi) {
            v2f b = *(const v2f*)(&sW[base + 32 * pi]);   // ds_load_b64
            acc[0][pi] = __builtin_amdgcn_wmma_f32_16x16x4_f32(false, a0, false, b, (short)0, acc[0][pi], false, false);
            acc[1][pi] = __builtin_amdgcn_wmma_f32_16x16x4_f32(false, a1, false, b, (short)0, acc[1][pi], false, false);
        }
    }

    // C/D layout: vgpr r, lanes 0-15 -> row +r, lanes 16-31 -> row +r+8
    const int rbase = m0 + ((lane >> 4) << 3);
    const int pc    = p0 + l15;
    float bb[4] = { bB[pc], bB[pc + 16], bB[pc + 32], bB[pc + 48] };
#pragma unroll
    for (int mi = 0; mi < 2; ++mi) {
#pragma unroll
        for (int r = 0; r < 8; ++r) {
            float* orow = Bu + (size_t)(rbase + 16 * mi + r) * Q + pc;
#pragma unroll
            for (int pi = 0; pi < 4; ++pi)
                orow[16 * pi] = acc[mi][pi][r] + bb[pi];
        }
    }
}

// ---------------------------------------------------------------------------
// Diagonal-A IM scan.  One lane per (b,q) channel; 4096 channels total.
//   s = 1/(1+dt^2 a);  x' = s x - dt s a z + dt s Bu;  z' = dt s x + s z + dt^2 s Bu
// ---------------------------------------------------------------------------
__global__ __launch_bounds__(256) void scan_kernel(
    const float* __restrict__ Bu,   // (MTOT, Q)
    const float* __restrict__ adiag,
    float* __restrict__ Y)          // (MTOT, Q)
{
    const int t  = blockIdx.x * 256 + threadIdx.x;   // 0..4095
    const int b  = t >> 8;
    const int qi = t & 255;

    const float dt  = 1.0f / (float)NSEQ;
    const float av  = adiag[qi];
    const float s   = 1.0f / (1.0f + dt * dt * av);
    const float dsa = dt * s * av;
    const float dss = dt * s;
    const float cfx = dt * s;
    const float cfz = dt * dt * s;

    const float* bu = Bu + (size_t)b * NSEQ * Q + qi;
    float*       y  = Y  + (size_t)b * NSEQ * Q + qi;

    float x = 0.0f, z = 0.0f;
    for (int n = 0; n < NSEQ; n += 4) {
        if (n + 16 < NSEQ)
            __builtin_prefetch(bu + (size_t)(n + 16) * Q, 0, 1);  // global_prefetch_b8
        float g0 = bu[(size_t)(n + 0) * Q];
        float g1 = bu[(size_t)(n + 1) * Q];
        float g2 = bu[(size_t)(n + 2) * Q];
        float g3 = bu[(size_t)(n + 3) * Q];

        float xn, zn;
        xn = s * x - dsa * z + cfx * g0;  zn = dss * x + s * z + cfz * g0;
        x = xn; z = zn; y[(size_t)(n + 0) * Q] = z;
        xn = s * x - dsa * z + cfx * g1;  zn = dss * x + s * z + cfz * g1;
        x = xn; z = zn; y[(size_t)(n + 1) * Q] = z;
        xn = s * x - dsa * z + cfx * g2;  zn = dss * x + s * z + cfz * g2;
        x = xn; z = zn; y[(size_t)(n + 2) * Q] = z;
        xn = s * x - dsa * z + cfx * g3;  zn = dss * x + s * z + cfz * g3;
        x = xn; z = zn; y[(size_t)(n + 3) * Q] = z;
    }
}

// ---------------------------------------------------------------------------
// GEMM 2 (dual) + epilogue:
//   x = Y@W_C^T + U@W_D^T + bC + bD ; g = gelu_exact(x) ; out = g*sigmoid(g) + U
// Stages both W_C and W_D chunks in LDS (160 KB total).
// ---------------------------------------------------------------------------
__global__ __launch_bounds__(256) void gemm_out_kernel(
    const float* __restrict__ Y,
    const float* __restrict__ U,
    const float* __restrict__ WC,   // (Q, Q) p-major
    const float* __restrict__ WD,   // (Q, Q) p-major
    const float* __restrict__ bC,
    const float* __restrict__ bD,
    float* __restrict__ Out)
{
    __shared__ float sC[(Q / 2) * PAIRSTR];
    __shared__ float sD[(Q / 2) * PAIRSTR];

    const int tid  = threadIdx.x;
    const int lane = tid & 31;
    const int wave = tid >> 5;
    const int m0   = (blockIdx.x * 8 + wave) * 32;
    const int p0   = blockIdx.y * 64;

    for (int i = 0; i < 64; ++i) {
        sC[wofs(tid, i)] = WC[(size_t)(p0 + i) * Q + tid];
        sD[wofs(tid, i)] = WD[(size_t)(p0 + i) * Q + tid];
    }
    __syncthreads();

    const int l15   = lane & 15;
    const int khalf = (lane >> 4) * 2;

    v8f acc[2][4];
#pragma unroll
    for (int mi = 0; mi < 2; ++mi)
#pragma unroll
        for (int pi = 0; pi < 4; ++pi) acc[mi][pi] = (v8f){};

    const float* yrow0 = Y + (size_t)(m0 + l15) * Q + khalf;
    const float* yrow1 = yrow0 + (size_t)16 * Q;
    const float* urow0 = U + (size_t)(m0 + l15) * Q + khalf;
    const float* urow1 = urow0 + (size_t)16 * Q;

    for (int k0 = 0; k0 < Q; k0 += 4) {
        v2f ay0 = *(const v2f*)(yrow0 + k0);
        v2f ay1 = *(const v2f*)(yrow1 + k0);
        v2f au0 = *(const v2f*)(urow0 + k0);
        v2f au1 = *(const v2f*)(urow1 + k0);
        const int base = wofs(k0 + khalf, l15);
#pragma unroll
        for (int pi = 0; pi < 4; ++pi) {
            v2f bc = *(const v2f*)(&sC[base + 32 * pi]);
            v2f bd = *(const v2f*)(&sD[base + 32 * pi]);
            acc[0][pi] = __builtin_amdgcn_wmma_f32_16x16x4_f32(false, ay0, false, bc, (short)0, acc[0][pi], false, false);
            acc[0][pi] = __builtin_amdgcn_wmma_f32_16x16x4_f32(false, au0, false, bd, (short)0, acc[0][pi], false, false);
            acc[1][pi] = __builtin_amdgcn_wmma_f32_16x16x4_f32(false, ay1, false, bc, (short)0, acc[1][pi], false, false);
            acc[1][pi] = __builtin_amdgcn_wmma_f32_16x16x4_f32(false, au1, false, bd, (short)0, acc[1][pi], false, false);
        }
    }

    const int rbase = m0 + ((lane >> 4) << 3);
    const int pc    = p0 + l15;
    float bias[4] = { bC[pc]      + bD[pc],
                      bC[pc + 16] + bD[pc + 16],
                      bC[pc + 32] + bD[pc + 32],
                      bC[pc + 48] + bD[pc + 48] };

#pragma unroll
    for (int mi = 0; mi < 2; ++mi) {
#pragma unroll
        for (int r = 0; r < 8; ++r) {
            const size_t row = (size_t)(rbase + 16 * mi + r) * Q + pc;
            const float* uin = U   + row;
            float*       out = Out + row;
#pragma unroll
            for (int pi = 0; pi < 4; ++pi) {
                float xv = acc[mi][pi][r] + bias[pi];
                float g  = 0.5f * xv * (1.0f + erff(xv * 0.70710678118654752f)); // exact GELU
                out[16 * pi] = g / (1.0f + expf(-g)) + uin[16 * pi];             // g*sigmoid(g)+u
            }
        }
    }
}

// ---------------------------------------------------------------------------
extern "C" void kernel_launch(void* const* d_in, const int* in_sizes, int n_in,
                              void* d_out, int out_size, void* d_ws, size_t ws_size,
                              hipStream_t stream) {
    (void)in_sizes; (void)n_in; (void)out_size; (void)d_ws; (void)ws_size;

    const float* u   = (const float*)d_in[0];
    const float* a   = (const float*)d_in[1];
    const float* W_B = (const float*)d_in[2];
    const float* b_B = (const float*)d_in[3];
    const float* W_C = (const float*)d_in[4];
    const float* b_C = (const float*)d_in[5];
    const float* W_D = (const float*)d_in[6];
    const float* b_D = (const float*)d_in[7];

    float* out_half = (float*)d_out;                         // final `out`, also temp Bu
    float* y_half   = (float*)d_out + (size_t)MTOT * Q;      // `y` output

    // 1) Bu = u @ W_B^T + b_B   (staged in the `out` half of d_out)
    gemm_bu_kernel<<<dim3(MTOT / 256, Q / 64), 256, 0, stream>>>(u, W_B, b_B, out_half);

    // 2) diagonal-A IM scan -> y
    scan_kernel<<<(BATCH * Q) / 256, 256, 0, stream>>>(out_half, a, y_half);

    // 3) out = GLU(GELU(y@W_C^T + u@W_D^T + bC + bD)) + u   (overwrites Bu staging)
    gemm_out_kernel<<<dim3(MTOT / 256, Q / 64), 256, 0, stream>>>(
        y_half, u, W_C, W_D, b_C, b_D, out_half);
}